// PhysicsCNN_21827023798596
// MI455X (gfx1250) — compile-verified
//
#include <hip/hip_runtime.h>

// ---------------------------------------------------------------------------
// PhysicsCNN for MI455X (gfx1250, wave32).
// Passes: zero-ws -> stats1 -> finalize1 -> stats2 -> finalize2 -> final(WMMA)
// ws float layout:
//   [0..15]    sum1        [16..31]  sumsq1
//   [32..63]   sum2        [64..95]  sumsq2
//   [96..111]  scale1      [112..127] shift1
//   [128..159] scale2      [160..191] shift2
// ---------------------------------------------------------------------------

typedef __attribute__((ext_vector_type(2))) float v2f;
typedef __attribute__((ext_vector_type(8))) float v8f;

#define BN_EPS 1e-5f

__device__ __forceinline__ float reluf(float a) { return fmaxf(a, 0.0f); }

__device__ __forceinline__ float waveReduceAdd(float v) {
  v += __shfl_xor(v, 16, 32);
  v += __shfl_xor(v, 8, 32);
  v += __shfl_xor(v, 4, 32);
  v += __shfl_xor(v, 2, 32);
  v += __shfl_xor(v, 1, 32);
  return v;
}

// conv1 (Cin=1, K=3, pad=1, L=3) + bias + relu for channel c
__device__ __forceinline__ void conv1_ch(float x0, float x1, float x2,
                                         const float* __restrict__ w1,
                                         const float* __restrict__ b1, int c,
                                         float& h0, float& h1, float& h2) {
  float wa = w1[3 * c + 0], wb = w1[3 * c + 1], wc = w1[3 * c + 2];
  float bb = b1[c];
  h0 = reluf(fmaf(wb, x0, fmaf(wc, x1, bb)));
  h1 = reluf(fmaf(wa, x0, fmaf(wb, x1, fmaf(wc, x2, bb))));
  h2 = reluf(fmaf(wa, x1, fmaf(wb, x2, bb)));
}

// grouped conv2 (depth pair) + bias + relu for out-channel o over h1n[c=o/2]
__device__ __forceinline__ void conv2_ch(float h0, float h1, float h2,
                                         const float* __restrict__ w2,
                                         const float* __restrict__ b2, int o,
                                         float& p0, float& p1, float& p2) {
  float ua = w2[3 * o + 0], ub = w2[3 * o + 1], uc = w2[3 * o + 2];
  float bb = b2[o];
  p0 = reluf(fmaf(ub, h0, fmaf(uc, h1, bb)));
  p1 = reluf(fmaf(ua, h0, fmaf(ub, h1, fmaf(uc, h2, bb))));
  p2 = reluf(fmaf(ua, h1, fmaf(ub, h2, bb)));
}

// ---------------------------------------------------------------------------
__global__ void zero_ws_kernel(float* __restrict__ ws, int n) {
  int i = blockIdx.x * blockDim.x + threadIdx.x;
  if (i < n) ws[i] = 0.0f;
}

// ---------------------------------------------------------------------------
__global__ __launch_bounds__(256) void stats1_kernel(
    const float* __restrict__ x, const float* __restrict__ w1,
    const float* __restrict__ b1, float* __restrict__ ws, int B) {
  int gtid = blockIdx.x * blockDim.x + threadIdx.x;
  int gstride = gridDim.x * blockDim.x;
  float s[16], q[16];
#pragma unroll
  for (int c = 0; c < 16; ++c) { s[c] = 0.0f; q[c] = 0.0f; }

  for (int i = gtid; i < B; i += gstride) {
    float x0 = x[3 * i + 0], x1 = x[3 * i + 1], x2 = x[3 * i + 2];
#pragma unroll
    for (int c = 0; c < 16; ++c) {
      float h0, h1, h2;
      conv1_ch(x0, x1, x2, w1, b1, c, h0, h1, h2);
      s[c] += h0 + h1 + h2;
      q[c] = fmaf(h0, h0, fmaf(h1, h1, fmaf(h2, h2, q[c])));
    }
  }

  __shared__ float acc[32];
  if (threadIdx.x < 32) acc[threadIdx.x] = 0.0f;
  __syncthreads();
  int lane = threadIdx.x & 31;
#pragma unroll
  for (int c = 0; c < 16; ++c) {
    float rs = waveReduceAdd(s[c]);
    float rq = waveReduceAdd(q[c]);
    if (lane == 0) {
      atomicAdd(&acc[c], rs);
      atomicAdd(&acc[16 + c], rq);
    }
  }
  __syncthreads();
  if (threadIdx.x < 32) atomicAdd(&ws[threadIdx.x], acc[threadIdx.x]);
}

// ---------------------------------------------------------------------------
__global__ void finalize1_kernel(const float* __restrict__ g1,
                                 const float* __restrict__ bt1,
                                 float* __restrict__ ws, int B) {
  int c = threadIdx.x;
  if (c >= 16) return;
  float n = 3.0f * (float)B;
  float mean = ws[c] / n;
  float var = ws[16 + c] / n - mean * mean;
  float sc = g1[c] * rsqrtf(var + BN_EPS);
  ws[96 + c] = sc;
  ws[112 + c] = fmaf(-mean, sc, bt1[c]);
}

// ---------------------------------------------------------------------------
__global__ __launch_bounds__(256) void stats2_kernel(
    const float* __restrict__ x, const float* __restrict__ w1,
    const float* __restrict__ b1, const float* __restrict__ w2,
    const float* __restrict__ b2, float* __restrict__ ws, int B) {
  const float* scale1 = ws + 96;
  const float* shift1 = ws + 112;
  int gtid = blockIdx.x * blockDim.x + threadIdx.x;
  int gstride = gridDim.x * blockDim.x;
  float s[32], q[32];
#pragma unroll
  for (int o = 0; o < 32; ++o) { s[o] = 0.0f; q[o] = 0.0f; }

  for (int i = gtid; i < B; i += gstride) {
    float x0 = x[3 * i + 0], x1 = x[3 * i + 1], x2 = x[3 * i + 2];
#pragma unroll
    for (int c = 0; c < 16; ++c) {
      float h0, h1, h2;
      conv1_ch(x0, x1, x2, w1, b1, c, h0, h1, h2);
      float sc = scale1[c], sh = shift1[c];
      h0 = fmaf(h0, sc, sh);
      h1 = fmaf(h1, sc, sh);
      h2 = fmaf(h2, sc, sh);
#pragma unroll
      for (int d = 0; d < 2; ++d) {
        int o = 2 * c + d;
        float p0, p1, p2;
        conv2_ch(h0, h1, h2, w2, b2, o, p0, p1, p2);
        s[o] += p0 + p1 + p2;
        q[o] = fmaf(p0, p0, fmaf(p1, p1, fmaf(p2, p2, q[o])));
      }
    }
  }

  __shared__ float acc[64];
  if (threadIdx.x < 64) acc[threadIdx.x] = 0.0f;
  __syncthreads();
  int lane = threadIdx.x & 31;
#pragma unroll
  for (int o = 0; o < 32; ++o) {
    float rs = waveReduceAdd(s[o]);
    float rq = waveReduceAdd(q[o]);
    if (lane == 0) {
      atomicAdd(&acc[o], rs);
      atomicAdd(&acc[32 + o], rq);
    }
  }
  __syncthreads();
  if (threadIdx.x < 64) atomicAdd(&ws[32 + threadIdx.x], acc[threadIdx.x]);
}

// ---------------------------------------------------------------------------
__global__ void finalize2_kernel(const float* __restrict__ g2,
                                 const float* __restrict__ bt2,
                                 float* __restrict__ ws, int B) {
  int c = threadIdx.x;
  if (c >= 32) return;
  float n = 3.0f * (float)B;
  float mean = ws[32 + c] / n;
  float var = ws[64 + c] / n - mean * mean;
  float sc = g2[c] * rsqrtf(var + BN_EPS);
  ws[128 + c] = sc;
  ws[160 + c] = fmaf(-mean, sc, bt2[c]);
}

// ---------------------------------------------------------------------------
// Final pass: recompute v[32] per sample, then head via V_WMMA_F32_16X16X4_F32.
// Wave handles 32 samples (one per lane) -> 2 tiles of M=16 x N=16, K=32 in
// eight k=4 steps. A-frag layout (f32 16x4): lanes 0-15 rows, half-wave
// selects K pair {k,k+1} vs {k+2,k+3}. C preloaded with fb1 broadcast.
// ---------------------------------------------------------------------------
__global__ __launch_bounds__(256) void final_kernel(
    const float* __restrict__ x, const float* __restrict__ w1,
    const float* __restrict__ b1, const float* __restrict__ w2,
    const float* __restrict__ b2, const float* __restrict__ fw1,
    const float* __restrict__ fb1, const float* __restrict__ fw2,
    const float* __restrict__ fb2, const float* __restrict__ ws,
    float* __restrict__ out) {
  constexpr int STR = 33;  // padded row stride (gcd(33,64)=1 -> conflict-free)
  __shared__ float vbuf[8 * 32 * STR];

  const float* scale1 = ws + 96;
  const float* shift1 = ws + 112;
  const float* scale2 = ws + 128;
  const float* shift2 = ws + 160;

  int tid = threadIdx.x;
  int wave = tid >> 5;
  int lane = tid & 31;
  int sample = blockIdx.x * 256 + tid;  // grid launched with exact full blocks

  float x0 = x[3 * sample + 0], x1 = x[3 * sample + 1], x2 = x[3 * sample + 2];

  float* row = &vbuf[(wave * 32 + lane) * STR];
#pragma unroll
  for (int c = 0; c < 16; ++c) {
    float h0, h1, h2;
    conv1_ch(x0, x1, x2, w1, b1, c, h0, h1, h2);
    float sc = scale1[c], sh = shift1[c];
    h0 = fmaf(h0, sc, sh);
    h1 = fmaf(h1, sc, sh);
    h2 = fmaf(h2, sc, sh);
#pragma unroll
    for (int d = 0; d < 2; ++d) {
      int o = 2 * c + d;
      float p0, p1, p2;
      conv2_ch(h0, h1, h2, w2, b2, o, p0, p1, p2);
      // temporal mean + folded batchnorm2
      row[o] = fmaf((p0 + p1 + p2), scale2[o] * (1.0f / 3.0f), shift2[o]);
    }
  }
  __syncthreads();

  int half = lane >> 4;  // K sub-pair select
  int lr = lane & 15;    // row (A) / col (B,D) index

  // B fragments: B[k][n] = fw1[n][k]; fw1 is (16 out, 32 in) row-major.
  v2f bfrag[8];
#pragma unroll
  for (int kk = 0; kk < 8; ++kk) {
    int k0 = kk * 4 + half * 2;
    bfrag[kk].x = fw1[lr * 32 + k0];
    bfrag[kk].y = fw1[lr * 32 + k0 + 1];
  }
  float bias1 = fb1[lr];
  float wv2 = fw2[lr];
  float bias2 = fb2[0];

  const float* wbase = &vbuf[wave * 32 * STR];
#pragma unroll
  for (int t = 0; t < 2; ++t) {
    v8f acc;
#pragma unroll
    for (int j = 0; j < 8; ++j) acc[j] = bias1;  // D starts at fb1[n]
#pragma unroll
    for (int kk = 0; kk < 8; ++kk) {
      v2f a;
      int r = t * 16 + lr;
      int k0 = kk * 4 + half * 2;
      a.x = wbase[r * STR + k0];
      a.y = wbase[r * STR + k0 + 1];
      acc = __builtin_amdgcn_wmma_f32_16x16x4_f32(
          /*neg_a=*/false, a, /*neg_b=*/false, bfrag[kk],
          /*c_mod=*/(short)0, acc, /*reuse_a=*/false, /*reuse_b=*/false);
    }
    // fc2: out[m] = fb2 + sum_n relu(D[m][n]) * fw2[n]; 16-lane butterfly.
    int obase = blockIdx.x * 256 + wave * 32 + t * 16 + half * 8;
#pragma unroll
    for (int j = 0; j < 8; ++j) {
      float r2 = reluf(acc[j]) * wv2;
      r2 += __shfl_xor(r2, 8, 32);
      r2 += __shfl_xor(r2, 4, 32);
      r2 += __shfl_xor(r2, 2, 32);
      r2 += __shfl_xor(r2, 1, 32);
      if (lr == 0) out[obase + j] = r2 + bias2;
    }
  }
}

// Scalar tail for B % 256 != 0 (keeps final_kernel's EXEC-all-ones invariant).
__global__ void tail_kernel(const float* __restrict__ x,
                            const float* __restrict__ w1,
                            const float* __restrict__ b1,
                            const float* __restrict__ w2,
                            const float* __restrict__ b2,
                            const float* __restrict__ fw1,
                            const float* __restrict__ fb1,
                            const float* __restrict__ fw2,
                            const float* __restrict__ fb2,
                            const float* __restrict__ ws,
                            float* __restrict__ out, int start, int B) {
  int i = start + blockIdx.x * blockDim.x + threadIdx.x;
  if (i >= B) return;
  const float* scale1 = ws + 96;
  const float* shift1 = ws + 112;
  const float* scale2 = ws + 128;
  const float* shift2 = ws + 160;
  float x0 = x[3 * i], x1 = x[3 * i + 1], x2 = x[3 * i + 2];
  float v[32];
#pragma unroll
  for (int c = 0; c < 16; ++c) {
    float h0, h1, h2;
    conv1_ch(x0, x1, x2, w1, b1, c, h0, h1, h2);
    float sc = scale1[c], sh = shift1[c];
    h0 = fmaf(h0, sc, sh);
    h1 = fmaf(h1, sc, sh);
    h2 = fmaf(h2, sc, sh);
#pragma unroll
    for (int d = 0; d < 2; ++d) {
      int o = 2 * c + d;
      float p0, p1, p2;
      conv2_ch(h0, h1, h2, w2, b2, o, p0, p1, p2);
      v[o] = fmaf((p0 + p1 + p2), scale2[o] * (1.0f / 3.0f), shift2[o]);
    }
  }
  float res = fb2[0];
#pragma unroll
  for (int n = 0; n < 16; ++n) {
    float h = fb1[n];
#pragma unroll
    for (int k = 0; k < 32; ++k) h = fmaf(v[k], fw1[n * 32 + k], h);
    res = fmaf(reluf(h), fw2[n], res);
  }
  out[i] = res;
}

// ---------------------------------------------------------------------------
extern "C" void kernel_launch(void* const* d_in, const int* in_sizes, int n_in,
                              void* d_out, int out_size, void* d_ws,
                              size_t ws_size, hipStream_t stream) {
  const float* x   = (const float*)d_in[0];
  const float* w1  = (const float*)d_in[1];
  const float* b1  = (const float*)d_in[2];
  const float* g1  = (const float*)d_in[3];
  const float* bt1 = (const float*)d_in[4];
  const float* w2  = (const float*)d_in[5];
  const float* b2  = (const float*)d_in[6];
  const float* g2  = (const float*)d_in[7];
  const float* bt2 = (const float*)d_in[8];
  const float* fw1 = (const float*)d_in[9];
  const float* fb1 = (const float*)d_in[10];
  const float* fw2 = (const float*)d_in[11];
  const float* fb2 = (const float*)d_in[12];
  float* out = (float*)d_out;
  float* ws = (float*)d_ws;

  int B = in_sizes[0] / 3;

  // Re-zero accumulators every launch (graph replays must be deterministic).
  zero_ws_kernel<<<1, 128, 0, stream>>>(ws, 96);

  int rb = (B + 255) / 256;
  if (rb > 1024) rb = 1024;  // cap same-address atomic depth ~1K
  if (rb < 1) rb = 1;

  stats1_kernel<<<rb, 256, 0, stream>>>(x, w1, b1, ws, B);
  finalize1_kernel<<<1, 16, 0, stream>>>(g1, bt1, ws, B);
  stats2_kernel<<<rb, 256, 0, stream>>>(x, w1, b1, w2, b2, ws, B);
  finalize2_kernel<<<1, 32, 0, stream>>>(g2, bt2, ws, B);

  int full = B / 256;
  if (full > 0)
    final_kernel<<<full, 256, 0, stream>>>(x, w1, b1, w2, b2, fw1, fb1, fw2,
                                           fb2, ws, out);
  int rem = B - full * 256;
  if (rem > 0)
    tail_kernel<<<(rem + 255) / 256, 256, 0, stream>>>(
        x, w1, b1, w2, b2, fw1, fb1, fw2, fb2, ws, out, full * 256, B);
}